// UVFeatureFusion_15951508537412
// MI455X (gfx1250) — compile-verified
//
#include <hip/hip_runtime.h>
#include <stdint.h>

// Problem constants (fixed by the reference)
#define B_   2
#define V_   8
#define C_   64
#define HW_  65536          // 256*256
#define A_   512
#define AA_  (A_*A_)        // 262144
#define IMGS (B_*V_)        // 16

// Tiling for the TDM-staged scatter
#define TILE_PIX      128                           // pixels per TDM tile (x dim)
#define TILES_PER_BLK 8
#define BLKS_PER_IMG  (HW_ / (TILE_PIX*TILES_PER_BLK))  // 64

typedef unsigned int u32;
typedef __attribute__((ext_vector_type(4))) u32 v4u;
typedef __attribute__((ext_vector_type(8))) int v8i;
typedef __attribute__((ext_vector_type(4))) int v4i;

// ---------------------------------------------------------------------------
// Zero the output buffer (fused accumulators + counts region).
// ---------------------------------------------------------------------------
__global__ __launch_bounds__(256) void zero_out(float* __restrict__ p, int n) {
  int i = blockIdx.x * blockDim.x + threadIdx.x;
  if (i < n) p[i] = 0.0f;
}

// ---------------------------------------------------------------------------
// Issue one TDM 2D tile load: 64 channels x TILE_PIX pixels of f32,
// channel stride = HW_ elements, into LDS at lds_byte.
// D# layout per CDNA5 ISA §8.3-8.6. This toolchain exposes the 6-arg builtin:
//   (v4u g0, v8i g1, v4i g2, v4i g3, v8i g4, i32 cpol)
// ---------------------------------------------------------------------------
__device__ __forceinline__ void tdm_issue_tile(const float* gsrc, u32 lds_byte) {
  uint64_t ga = (uint64_t)(uintptr_t)gsrc;

  v4u g0;
  g0.x = 1u;                                   // count=1 valid descriptor
  g0.y = lds_byte;                             // lds_addr (bytes)
  g0.z = (u32)(ga & 0xFFFFFFFFu);              // global_addr[31:0]
  g0.w = (u32)((ga >> 32) & 0x01FFFFFFu)       // global_addr[56:32]
       | (2u << 30);                           // type=2 ("image")

  v8i g1;
  g1.s0 = 0x00020000;                          // data_size=2 (4 bytes); no iterate/pad
  g1.s1 = 0;                                   // barrier addr=0; tensor_dim0 lo16 (65536->0)
  g1.s2 = (64 << 16) | 1;                      // tensor_dim1 lo16=64 | tensor_dim0 hi16=1
  g1.s3 = (TILE_PIX << 16);                    // tile_dim0=128 | tensor_dim1 hi16=0
  g1.s4 = 64;                                  // tile_dim2=0 | tile_dim1=64
  g1.s5 = HW_;                                 // tensor_dim0_stride lo32 = 65536
  g1.s6 = 0;                                   // dim1_stride lo16 | dim0_stride hi16
  g1.s7 = 0;

  v4i gz4 = {0, 0, 0, 0};                      // groups 2/3 unused (2D tensor)
  v8i gz8 = {0, 0, 0, 0, 0, 0, 0, 0};         // extra group (unused)
  __builtin_amdgcn_tensor_load_to_lds(g0, g1, gz4, gz4, gz8, 0);
}

// ---------------------------------------------------------------------------
// Scatter kernel: one block = 8 TDM tiles of one (b,v) image.
// 256 threads: thread pair (p, p+128) shares pixel p, each half does 32
// channels of global_atomic_add_f32. Wave 0 drives the double-buffered TDM
// pipeline; s_wait_tensorcnt + barriers gate buffer reuse.
// ---------------------------------------------------------------------------
__global__ __launch_bounds__(256) void scatter_k(
    const float*  __restrict__ enc,   // [B,V,C,H,W]
    const float2* __restrict__ uvs,   // [B,V,H,W,2]
    const float*  __restrict__ msk,   // [B,V,1,H,W]
    const float*  __restrict__ imp,   // [B,V]
    float* __restrict__ out)          // fused [B,C,A,A] ++ counts [B,A,A]
{
  __shared__ float tile[2][C_ * TILE_PIX];     // 2 x 32KB double buffer

  const int tid    = threadIdx.x;
  const int img    = blockIdx.x / BLKS_PER_IMG;        // b*V + v
  const int chunk  = blockIdx.x % BLKS_PER_IMG;
  const int pixBase = chunk * (TILE_PIX * TILES_PER_BLK);
  const int b      = img / V_;

  const float wimg = fmaxf(imp[img], 0.0f);            // stop_gradient + relu
  const float* encImg = enc + (size_t)img * C_ * HW_;
  float* fused  = out;
  float* counts = out + (size_t)B_ * C_ * AA_;

  const int p     = tid & (TILE_PIX - 1);              // pixel within tile
  const int chalf = tid >> 7;                          // 0: c[0,32) 1: c[32,64)
  const bool wave0 = (tid < 32);                       // wave-uniform (wave32)

  if (wave0) {
    tdm_issue_tile(encImg + pixBase, (u32)(uintptr_t)&tile[0][0]);
  }

  for (int t = 0; t < TILES_PER_BLK; ++t) {
    if (wave0) {
      if (t + 1 < TILES_PER_BLK) {
        // Buffer (t+1)&1 was released by the barrier ending iteration t-1.
        tdm_issue_tile(encImg + pixBase + (t + 1) * TILE_PIX,
                       (u32)(uintptr_t)&tile[(t + 1) & 1][0]);
        __builtin_amdgcn_s_wait_tensorcnt(1);   // tile t complete, t+1 in flight
      } else {
        __builtin_amdgcn_s_wait_tensorcnt(0);   // last tile complete
      }
    }
    __syncthreads();                            // LDS tile t visible to all waves

    const int pix = pixBase + t * TILE_PIX + p;
    const float2 uv = uvs[(size_t)img * HW_ + pix];
    const float  m  = msk[(size_t)img * HW_ + pix];

    float w = ((m > 0.5f) && __builtin_isfinite(uv.x) && __builtin_isfinite(uv.y))
                ? wimg : 0.0f;
    if (w > 0.0f) {
      float u = fminf(fmaxf(uv.x, 0.0f), 1.0f);
      float v = fminf(fmaxf(uv.y, 0.0f), 1.0f);
      int xi = (int)rintf(u * (float)(A_ - 1));            // round-half-even
      int yi = (int)rintf((1.0f - v) * (float)(A_ - 1));
      int idx = yi * A_ + xi;

      const float* src  = &tile[t & 1][0];
      float* dstB = fused + (size_t)b * C_ * AA_ + idx;
      const int c0 = chalf << 5;
#pragma unroll
      for (int cc = 0; cc < 32; ++cc) {
        const int c = c0 + cc;
        atomicAdd(dstB + (size_t)c * AA_, w * src[(c << 7) | p]);
      }
      if (chalf == 0) {
        atomicAdd(counts + (size_t)b * AA_ + idx, w);
      }
    }
    __syncthreads();                            // release buffer t&1 for reuse
  }
}

// ---------------------------------------------------------------------------
// Normalize: fused /= max(counts,1); validity = counts > 0 (in place).
// One thread per atlas cell; channel loop strided by AA_ stays coalesced.
// ---------------------------------------------------------------------------
__global__ __launch_bounds__(256) void norm_k(float* __restrict__ out) {
  const int t = blockIdx.x * blockDim.x + threadIdx.x;   // [0, B*AA)
  if (t >= B_ * AA_) return;
  float* fused  = out;
  float* counts = out + (size_t)B_ * C_ * AA_;

  const int b    = t / AA_;
  const int cell = t - b * AA_;
  const float cnt = counts[t];
  const float d   = fmaxf(cnt, 1.0f);

  float* f = fused + (size_t)b * C_ * AA_ + cell;
#pragma unroll 8
  for (int c = 0; c < C_; ++c) {
    f[(size_t)c * AA_] = f[(size_t)c * AA_] / d;
  }
  counts[t] = (cnt > 0.0f) ? 1.0f : 0.0f;               // validity output
}

// ---------------------------------------------------------------------------
extern "C" void kernel_launch(void* const* d_in, const int* in_sizes, int n_in,
                              void* d_out, int out_size, void* d_ws, size_t ws_size,
                              hipStream_t stream) {
  (void)in_sizes; (void)n_in; (void)d_ws; (void)ws_size;

  const float*  enc = (const float*)d_in[0];   // encoded_views [2,8,64,256,256]
  const float2* uvs = (const float2*)d_in[1];  // view_uvs      [2,8,256,256,2]
  const float*  msk = (const float*)d_in[2];   // view_masks    [2,8,1,256,256]
  const float*  imp = (const float*)d_in[3];   // view_importance [2,8]
  // d_in[4] = atlas_size scalar (512, hardcoded)

  float* out = (float*)d_out;                  // fused (33.5M) ++ validity (0.5M)

  zero_out<<<(out_size + 255) / 256, 256, 0, stream>>>(out, out_size);

  scatter_k<<<IMGS * BLKS_PER_IMG, 256, 0, stream>>>(enc, uvs, msk, imp, out);

  norm_k<<<(B_ * AA_ + 255) / 256, 256, 0, stream>>>(out);
}